// MACE_openmm_59193239274052
// MI455X (gfx1250) — compile-verified
//
#include <hip/hip_runtime.h>
#include <math.h>

// MACE edge-energy surrogate for MI455X (gfx1250, wave32, WMMA).
//
//   wave  <-> one (shift s, receiver i)                 (27*N waves)
//   tile  <-> 16 sender atoms j per pass
//   lane p = lane&15 -> pair slot M (A row) / hidden n (B, D column); half = lane>>4 -> K-half
//
// Layer 1 (feat[16x8] @ W1[8x16] + b1): two chained V_WMMA_F32_16X16X4_F32.
// Layer 2 (h[16x16] @ W2[16] + b2):     SiLU + per-lane masked accumulate
//   using a wave-uniform ballot keep-mask (no per-tile cross-lane traffic);
//   b2 recovered exactly as b2 * popc(keep) per tile.
// Tiles with an all-zero keep mask take a uniform s_cbranch past all the
// transcendental + WMMA work (>99% of tiles at this density/cutoff), which
// is bit-identical to the reference since masked edges contribute 0.

typedef float v2f __attribute__((ext_vector_type(2)));
typedef float v8f __attribute__((ext_vector_type(8)));

#define NMAX   512
#define RMAX   5.0f
#define PI_F   3.14159265358979323846f

__global__ void zero_out_kernel(float* out) { out[0] = 0.0f; }

__global__ __launch_bounds__(256) void mace_edge_energy(
    const float* __restrict__ pos_nm,   // [N,3]
    const float* __restrict__ box_nm,   // [3,3]
    const float* __restrict__ W1,       // [8,16] row-major (basis k, hidden n)
    const float* __restrict__ b1,       // [16]
    const float* __restrict__ W2,       // [16]
    const float* __restrict__ b2,       // [1]
    float* __restrict__ out,            // [1]
    int N)
{
    __shared__ float spos[NMAX * 3];

    // Stage scaled positions (nm -> Angstrom, x10) once per block.
    for (int t = threadIdx.x; t < N * 3; t += blockDim.x)
        spos[t] = pos_nm[t] * 10.0f;
    __syncthreads();

    const int lane = threadIdx.x & 31;
    const int wave = (int)((blockIdx.x * blockDim.x + threadIdx.x) >> 5);
    if (wave >= 27 * N) return;

    const int s = wave / N;   // shift image 0..26
    const int i = wave % N;   // receiver atom

    // S[s] = (s/9-1, (s/3)%3-1, s%3-1) @ cell, cell = box*10
    const int sx = (s / 9)       - 1;
    const int sy = ((s / 3) % 3) - 1;
    const int sz = (s % 3)       - 1;
    float cell[9];
#pragma unroll
    for (int t = 0; t < 9; ++t) cell[t] = box_nm[t] * 10.0f;
    const float shx = sx * cell[0] + sy * cell[3] + sz * cell[6];
    const float shy = sx * cell[1] + sy * cell[4] + sz * cell[7];
    const float shz = sx * cell[2] + sy * cell[5] + sz * cell[8];
    const bool  zero_shift = (s == 13);

    // diff = pos_j - pos_i + shift = pos_j - c, with c folded once.
    const float cx = spos[i * 3 + 0] - shx;
    const float cy = spos[i * 3 + 1] - shy;
    const float cz = spos[i * 3 + 2] - shz;

    const int   p    = lane & 15;   // pair slot M; hidden column n for B/C/D
    const int   half = lane >> 4;   // K-half
    const float w    = PI_F / RMAX;
    const float pref = 0.6324555320336759f;  // sqrt(2/R_MAX)

    // Bessel angular coefficients this lane needs: n = 2*half + {1,2,5,6}.
    const float fa = (float)(2 * half + 1) * w;
    const float fb = (float)(2 * half + 2) * w;
    const float fc = (float)(2 * half + 5) * w;
    const float fd = (float)(2 * half + 6) * w;

    // B operands (W1 rows = K, cols = n): VGPR v, lane-half h -> K = 2*h + v.
    v2f B1, B2;
    B1.x = W1[(2 * half + 0) * 16 + p];
    B1.y = W1[(2 * half + 1) * 16 + p];
    B2.x = W1[(2 * half + 4) * 16 + p];
    B2.y = W1[(2 * half + 5) * 16 + p];

    // C accumulator pre-loaded with b1[n] (C/D element (r,lane) has N=lane%16).
    const float b1n = b1[p];
    v8f Cb;
#pragma unroll
    for (int t = 0; t < 8; ++t) Cb[t] = b1n;

    const float w2n = W2[p];
    const float b2v = b2[0];

    float        acc  = 0.0f;  // sum of h@W2 over kept edges (per lane partial)
    unsigned int ecnt = 0u;    // kept-edge count (wave-uniform, lives in SGPR)

    for (int j0 = 0; j0 < N; j0 += 16) {
        const int jj = j0 + p;
        const int j  = (jj < N) ? jj : (N - 1);

        const float dx = spos[j * 3 + 0] - cx;
        const float dy = spos[j * 3 + 1] - cy;
        const float dz = spos[j * 3 + 2] - cz;
        const float r2 = dx * dx + dy * dy + dz * dz;

        const bool keep = (r2 < RMAX * RMAX) && !(zero_shift && (i == j)) && (jj < N);

        // Wave-uniform keep mask: bit M = keep of pair M (halves duplicate).
        const unsigned int kmask =
            __builtin_amdgcn_ballot_w32(keep) & 0xFFFFu;
        ecnt += (unsigned int)__popc(kmask);

        if (kmask) {   // uniform branch: EXEC stays all-1s for the WMMAs
            const float r    = __builtin_sqrtf(keep ? r2 : 1.0f);
            const float fcut = 0.5f * (__cosf(w * r) + 1.0f);
            const float sc   = pref * fcut * __builtin_amdgcn_rcpf(r);

            // A tile: lane (M=p, half) supplies feat[p,K], K = 2*half+v (+4).
            v2f A1, A2;
            A1.x = sc * __sinf(fa * r);
            A1.y = sc * __sinf(fb * r);
            A2.x = sc * __sinf(fc * r);
            A2.y = sc * __sinf(fd * r);

            // h_pre = feat @ W1 + b1 via two chained fp32 WMMAs.
            v8f D = __builtin_amdgcn_wmma_f32_16x16x4_f32(
                        false, A1, false, B1, (short)0, Cb, false, false);
            D = __builtin_amdgcn_wmma_f32_16x16x4_f32(
                        false, A2, false, B2, (short)0, D, false, false);

            // SiLU + W2 + masked local accumulate.
            // D element (rr,lane): pair M = rr + 8*half, hidden N = p.
#pragma unroll
            for (int rr = 0; rr < 8; ++rr) {
                const float hp = D[rr];
                const float sg = __builtin_amdgcn_rcpf(1.0f + __expf(-hp));
                const float t  = hp * sg * w2n;
                const unsigned int bit = (kmask >> (rr + 8 * half)) & 1u;
                acc += bit ? t : 0.0f;
            }
        }
    }

    // Single wave-level reduction at the end; one atomic per wave.
    acc += __shfl_xor(acc, 1, 32);
    acc += __shfl_xor(acc, 2, 32);
    acc += __shfl_xor(acc, 4, 32);
    acc += __shfl_xor(acc, 8, 32);
    acc += __shfl_xor(acc, 16, 32);
    if (lane == 0)
        atomicAdd(out, acc + b2v * (float)ecnt);
}

extern "C" void kernel_launch(void* const* d_in, const int* in_sizes, int n_in,
                              void* d_out, int out_size, void* d_ws, size_t ws_size,
                              hipStream_t stream) {
    const float* pos = (const float*)d_in[0];
    const float* box = (const float*)d_in[1];
    const float* W1  = (const float*)d_in[2];
    const float* b1  = (const float*)d_in[3];
    const float* W2  = (const float*)d_in[4];
    const float* b2  = (const float*)d_in[5];
    float* out = (float*)d_out;

    const int N = in_sizes[0] / 3;          // 512
    zero_out_kernel<<<1, 1, 0, stream>>>(out);

    const long long nwaves  = 27LL * N;     // one wave per (shift, i)
    const long long threads = nwaves * 32;
    const int blocks = (int)((threads + 255) / 256);
    mace_edge_energy<<<blocks, 256, 0, stream>>>(pos, box, W1, b1, W2, b2, out, N);
}